// BEM_50002009260181
// MI455X (gfx1250) — compile-verified
//
#include <hip/hip_runtime.h>
#include <math.h>

typedef __attribute__((ext_vector_type(2))) float v2f;
typedef __attribute__((ext_vector_type(8))) float v8f;

#define BATCH   4
#define LSEQ    1024
#define DDIM    768
#define FDIM    32
#define NROWS   (BATCH * LSEQ)      // 4096
#define NTILE_M (NROWS / 16)        // 256
#define NTILE_D (DDIM / 16)         // 48

// ---------------------------------------------------------------------------
// Branch-free transcendentals on the gfx1250 TRANS pipe.
// ---------------------------------------------------------------------------
__device__ __forceinline__ float fast_tanh(float x) {
#if __has_builtin(__builtin_amdgcn_tanhf)
    return __builtin_amdgcn_tanhf(x);             // v_tanh_f32
#elif __has_builtin(__builtin_amdgcn_tanh_f32)
    return __builtin_amdgcn_tanh_f32(x);          // v_tanh_f32
#else
    // tanh(x) = sign(x) * (1 - e^-2|x|) / (1 + e^-2|x|), e^-2|x| = 2^(-2|x|*log2e)
    const float ax = __builtin_fabsf(x);
    const float t  = __builtin_amdgcn_exp2f(ax * -2.885390081777927f);
    const float r  = (1.0f - t) * __builtin_amdgcn_rcpf(1.0f + t);
    return __builtin_copysignf(r, x);
#endif
}

__device__ __forceinline__ float fast_tan(float x) {
    // v_sin/v_cos take revolutions (ocml native_sin scaling); args stay < 1 rev.
    const float r2pi = 0.15915494309189535f;
    const float s = __builtin_amdgcn_sinf(x * r2pi);
    const float c = __builtin_amdgcn_cosf(x * r2pi);
    return s * __builtin_amdgcn_rcpf(c);
}

__device__ __forceinline__ void load_wfrag(const float* __restrict__ W, int d,
                                           int kh, v2f f[8]) {
    #pragma unroll
    for (int k = 0; k < 8; ++k)
        f[k] = *(const v2f*)(W + (size_t)d * FDIM + 4 * k + 2 * kh);
}

// ---------------------------------------------------------------------------
// Kernel 1: fused key projections (f32 WMMA), T-driven query terms, and the
// Sum_d(TAQ*VKey) / Sum_d(TVQ*AKey) reductions. One block = one 16-row tile
// of (B*L); 4 waves split the 48 d-tiles. Weight fragments are software-
// pipelined one d-tile ahead so global-load latency hides under the WMMA chain.
// ---------------------------------------------------------------------------
__global__ __launch_bounds__(128)
void bem_scores(const float* __restrict__ T, const float* __restrict__ A,
                const float* __restrict__ V,
                const float* __restrict__ w_a, const float* __restrict__ b_a,
                const float* __restrict__ w_v, const float* __restrict__ b_v,
                const float* __restrict__ W_aup1, const float* __restrict__ b_aup1,
                const float* __restrict__ W_vup1, const float* __restrict__ b_vup1,
                float* __restrict__ scoreA, float* __restrict__ scoreV)
{
    __shared__ float sa[16], sv[16];
    __shared__ float redA[4][16], redV[4][16];

    const int row0 = blockIdx.x * 16;       // global row in flattened (B*L)
    const int tid  = threadIdx.x;
    const int wave = tid >> 5;
    const int tl   = tid & 31;
    const int ln   = tl & 15;               // M-lane (A frag) / N-lane (B,C frags)
    const int kh   = tl >> 4;               // lane half

    // per-row scalar projections s_a = A.w_a, s_v = V.w_v
    if (tid < 32) {
        const int r = tid & 15;
        const float* x = (tid < 16 ? A : V) + (size_t)(row0 + r) * FDIM;
        const float* w = (tid < 16 ? w_a : w_v);
        float s = 0.f;
        #pragma unroll
        for (int f = 0; f < FDIM; ++f) s = fmaf(x[f], w[f], s);
        if (tid < 16) sa[r] = s; else sv[r] = s;
    }
    __syncthreads();

    // Hoisted activation fragments, reused for all d-tiles.
    v2f fa[8], fv[8];
    #pragma unroll
    for (int k = 0; k < 8; ++k) {
        fa[k] = *(const v2f*)(A + (size_t)(row0 + ln) * FDIM + 4 * k + 2 * kh);
        fv[k] = *(const v2f*)(V + (size_t)(row0 + ln) * FDIM + 4 * k + 2 * kh);
    }

    float sarr[8], svarr[8];
    #pragma unroll
    for (int r = 0; r < 8; ++r) { sarr[r] = sa[r + 8 * kh]; svarr[r] = sv[r + 8 * kh]; }
    const float ba = b_a[0], bv = b_v[0];

    float accA[8] = {0.f, 0.f, 0.f, 0.f, 0.f, 0.f, 0.f, 0.f};
    float accV[8] = {0.f, 0.f, 0.f, 0.f, 0.f, 0.f, 0.f, 0.f};

    // Prologue: fragments + biases for this wave's first d-tile.
    v2f bAf[8], bVf[8];
    {
        const int d0 = wave * 16 + ln;
        load_wfrag(W_vup1, d0, kh, bVf);
        load_wfrag(W_aup1, d0, kh, bAf);
    }
    float biasV = b_vup1[wave * 16 + ln];
    float biasA = b_aup1[wave * 16 + ln];

    for (int dt = wave; dt < NTILE_D; dt += 4) {
        const int dcur = dt * 16 + ln;
        const int dtn  = (dt + 4 < NTILE_D) ? dt + 4 : dt;   // clamp, branch-free
        const int dnxt = dtn * 16 + ln;

        // Issue next tile's weight fragments + biases early.
        v2f nAf[8], nVf[8];
        load_wfrag(W_vup1, dnxt, kh, nVf);
        load_wfrag(W_aup1, dnxt, kh, nAf);
        const float nbV = b_vup1[dnxt];
        const float nbA = b_aup1[dnxt];

        // Issue this tile's T column loads early.
        float tld[8];
        #pragma unroll
        for (int r = 0; r < 8; ++r)
            tld[r] = T[(size_t)(row0 + r + 8 * kh) * DDIM + dcur];
        __builtin_prefetch(T + (size_t)(row0 + 8 * kh) * DDIM + dcur + 64, 0, 1);

        // 16-WMMA chain on the previously fetched fragments.
        v8f cA = {}, cV = {};
        #pragma unroll
        for (int k = 0; k < 8; ++k) {
            cV = __builtin_amdgcn_wmma_f32_16x16x4_f32(false, fv[k], false, bVf[k],
                                                       (short)0, cV, false, false);
            cA = __builtin_amdgcn_wmma_f32_16x16x4_f32(false, fa[k], false, bAf[k],
                                                       (short)0, cA, false, false);
        }

        // Branch-free epilogue on the TRANS pipe.
        #pragma unroll
        for (int r = 0; r < 8; ++r) {
            const float vkey = fast_tanh(cV[r] + biasV);
            const float akey = fast_tanh(cA[r] + biasA);
            accA[r] = fmaf(fast_tanh(fmaf(tld[r], sarr[r], ba)), vkey, accA[r]);
            accV[r] = fmaf(fast_tanh(fmaf(tld[r], svarr[r], bv)), akey, accV[r]);
        }

        // Rotate pipeline registers.
        #pragma unroll
        for (int k = 0; k < 8; ++k) { bVf[k] = nVf[k]; bAf[k] = nAf[k]; }
        biasV = nbV; biasA = nbA;
    }

    // Reduce across the 16 lanes of each half (masks < 16 never cross halves).
    #pragma unroll
    for (int r = 0; r < 8; ++r) {
        float a = accA[r], v = accV[r];
        #pragma unroll
        for (int off = 1; off < 16; off <<= 1) {
            a += __shfl_xor(a, off, 32);
            v += __shfl_xor(v, off, 32);
        }
        accA[r] = a; accV[r] = v;
    }
    if (ln == 0) {
        #pragma unroll
        for (int r = 0; r < 8; ++r) {
            redA[wave][kh * 8 + r] = accA[r];
            redV[wave][kh * 8 + r] = accV[r];
        }
    }
    __syncthreads();
    if (tid < 16) {
        scoreA[row0 + tid] = redA[0][tid] + redA[1][tid] + redA[2][tid] + redA[3][tid];
        scoreV[row0 + tid] = redV[0][tid] + redV[1][tid] + redV[2][tid] + redV[3][tid];
    }
}

// ---------------------------------------------------------------------------
// Kernel 2: softmax over L per batch. blockIdx = batch*2 + {score_a,score_v}.
// ---------------------------------------------------------------------------
__global__ __launch_bounds__(256)
void bem_softmax(const float* __restrict__ sc, float* __restrict__ wt)
{
    __shared__ float red[256];
    const int sel = blockIdx.x & 1;
    const int b   = blockIdx.x >> 1;
    const float* s = sc + sel * NROWS + b * LSEQ;
    float*       w = wt + sel * NROWS + b * LSEQ;
    const int t = threadIdx.x;

    float v[4];
    float m = -INFINITY;
    #pragma unroll
    for (int j = 0; j < 4; ++j) { v[j] = s[t + 256 * j]; m = fmaxf(m, v[j]); }
    red[t] = m; __syncthreads();
    for (int off = 128; off > 0; off >>= 1) {
        if (t < off) red[t] = fmaxf(red[t], red[t + off]);
        __syncthreads();
    }
    m = red[0]; __syncthreads();

    float sum = 0.f;
    #pragma unroll
    for (int j = 0; j < 4; ++j) {
        v[j] = __builtin_amdgcn_exp2f((v[j] - m) * 1.4426950408889634f);
        sum += v[j];
    }
    red[t] = sum; __syncthreads();
    for (int off = 128; off > 0; off >>= 1) {
        if (t < off) red[t] += red[t + off];
        __syncthreads();
    }
    const float inv = __builtin_amdgcn_rcpf(red[0]);
    #pragma unroll
    for (int j = 0; j < 4; ++j) w[t + 256 * j] = v[j] * inv;
}

// ---------------------------------------------------------------------------
// Kernel 3: value projections via WMMA + softmax scaling + store.
// One wave owns one 16x16 tile and produces both outputs for it.
// ---------------------------------------------------------------------------
__global__ __launch_bounds__(128)
void bem_out(const float* __restrict__ A, const float* __restrict__ V,
             const float* __restrict__ W_aup2, const float* __restrict__ b_aup2,
             const float* __restrict__ W_vup2, const float* __restrict__ b_vup2,
             const float* __restrict__ wt,   // [ta (4096) | tv (4096)]
             float* __restrict__ out)
{
    const int tid  = threadIdx.x;
    const int wave = tid >> 5;
    const int tl   = tid & 31;
    const int ln   = tl & 15;
    const int kh   = tl >> 4;
    const int gw   = blockIdx.x * 4 + wave;     // 0 .. 12287
    const int row0 = (gw / NTILE_D) * 16;
    const int d    = (gw % NTILE_D) * 16 + ln;

    // Issue every load up front: activations, weights, biases, softmax weights.
    v2f fa[8], fv[8], bAf[8], bVf[8];
    #pragma unroll
    for (int k = 0; k < 8; ++k) {
        fa[k] = *(const v2f*)(A + (size_t)(row0 + ln) * FDIM + 4 * k + 2 * kh);
        fv[k] = *(const v2f*)(V + (size_t)(row0 + ln) * FDIM + 4 * k + 2 * kh);
    }
    load_wfrag(W_aup2, d, kh, bAf);
    load_wfrag(W_vup2, d, kh, bVf);
    const float biasA = b_aup2[d], biasV = b_vup2[d];
    const float* ta = wt;
    const float* tv = wt + NROWS;
    float wa8[8], wv8[8];
    #pragma unroll
    for (int r = 0; r < 8; ++r) {
        wa8[r] = ta[row0 + r + 8 * kh];
        wv8[r] = tv[row0 + r + 8 * kh];
    }

    v8f cA = {}, cV = {};
    #pragma unroll
    for (int k = 0; k < 8; ++k) {
        cA = __builtin_amdgcn_wmma_f32_16x16x4_f32(false, fa[k], false, bAf[k],
                                                   (short)0, cA, false, false);
        cV = __builtin_amdgcn_wmma_f32_16x16x4_f32(false, fv[k], false, bVf[k],
                                                   (short)0, cV, false, false);
    }

    float* out0 = out;
    float* out1 = out + (size_t)NROWS * DDIM;
    #pragma unroll
    for (int r = 0; r < 8; ++r) {
        const int m = r + 8 * kh;
        const size_t idx = (size_t)(row0 + m) * DDIM + d;
        out0[idx] = fast_tan (cA[r] + biasA) * wa8[r];   // AValue uses tan (!)
        out1[idx] = fast_tanh(cV[r] + biasV) * wv8[r];   // VValue uses tanh
    }
}

// ---------------------------------------------------------------------------
extern "C" void kernel_launch(void* const* d_in, const int* in_sizes, int n_in,
                              void* d_out, int out_size, void* d_ws, size_t ws_size,
                              hipStream_t stream) {
    const float* T      = (const float*)d_in[0];
    const float* A      = (const float*)d_in[1];
    const float* V      = (const float*)d_in[2];
    const float* w_a    = (const float*)d_in[3];
    const float* b_a    = (const float*)d_in[4];
    const float* w_v    = (const float*)d_in[5];
    const float* b_v    = (const float*)d_in[6];
    const float* W_aup1 = (const float*)d_in[7];
    const float* b_aup1 = (const float*)d_in[8];
    const float* W_aup2 = (const float*)d_in[9];
    const float* b_aup2 = (const float*)d_in[10];
    const float* W_vup1 = (const float*)d_in[11];
    const float* b_vup1 = (const float*)d_in[12];
    const float* W_vup2 = (const float*)d_in[13];
    const float* b_vup2 = (const float*)d_in[14];

    float* out = (float*)d_out;
    float* ws  = (float*)d_ws;
    float* sc  = ws;                 // 2 * 4096 floats of raw scores
    float* wt  = ws + 2 * NROWS;     // 2 * 4096 floats of softmax weights

    bem_scores<<<NTILE_M, 128, 0, stream>>>(T, A, V, w_a, b_a, w_v, b_v,
                                            W_aup1, b_aup1, W_vup1, b_vup1,
                                            sc, sc + NROWS);
    bem_softmax<<<2 * BATCH, 256, 0, stream>>>(sc, wt);
    bem_out<<<(NTILE_M * NTILE_D) / 4, 128, 0, stream>>>(A, V, W_aup2, b_aup2,
                                                         W_vup2, b_vup2, wt, out);
}